// GATConv_68143951118620
// MI455X (gfx1250) — compile-verified
//
#include <hip/hip_runtime.h>

typedef __attribute__((ext_vector_type(2))) float v2f;
typedef __attribute__((ext_vector_type(8))) float v8f;

#define IN_C   128
#define OUT_C  128
#define HEADS  4
#define HEAD_D 32

// ---------------- init: zero out / amax(transformed) / asum --------------
__global__ void gat_zero(float* out, long long n_out,
                         unsigned* amax_u, long long n_amax,
                         float* asum, long long n_asum) {
  long long i = (long long)blockIdx.x * blockDim.x + threadIdx.x;
  long long stride = (long long)gridDim.x * blockDim.x;
  for (long long t = i; t < n_out;  t += stride) out[t] = 0.0f;
  for (long long t = i; t < n_amax; t += stride) amax_u[t] = 0u;   // 0 == "empty segment"
  for (long long t = i; t < n_asum; t += stride) asum[t] = 0.0f;
}

// ---------------- GEMM: xw = x @ W  via V_WMMA_F32_16X16X4_F32 -----------
// Block = 256 threads = 8 waves. Block computes a 16-row tile of xw;
// wave w computes columns [16w, 16w+16). K accumulated 4 at a time.
__global__ void __launch_bounds__(256) gat_gemm_xw(const float* __restrict__ x,
                                                   const float* __restrict__ W,
                                                   float* __restrict__ xw,
                                                   int N) {
  __shared__ float xA[16][IN_C];              // 8 KB staging of the A tile
  const int t = threadIdx.x;
  const int m_base = blockIdx.x * 16;

  // cooperative load of 16 x 128 floats (zero-pad rows past N)
  for (int i = t; i < 16 * (IN_C / 4); i += 256) {
    int r  = i / (IN_C / 4);
    int c4 = i % (IN_C / 4);
    int gr = m_base + r;
    float4 v = make_float4(0.f, 0.f, 0.f, 0.f);
    if (gr < N) v = ((const float4*)(x + (long long)gr * IN_C))[c4];
    ((float4*)&xA[r][0])[c4] = v;
  }
  __syncthreads();

  const int wave   = t >> 5;         // 0..7 -> 16-col tile of W / xw
  const int lane   = t & 31;
  const int half   = lane >> 4;      // 0: lanes 0-15, 1: lanes 16-31
  const int l      = lane & 15;
  const int n_base = wave * 16;

  v8f c = {};
  for (int k = 0; k < IN_C; k += 4) {
    // A 16x4 fp32 layout: lanes0-15 hold K=k,k+1 (M=lane); lanes16-31 hold K=k+2,k+3
    v2f a, b;
    a[0] = xA[l][k + 2 * half + 0];
    a[1] = xA[l][k + 2 * half + 1];
    // B 4x16 fp32, rows striped across lanes: VGPR0 = rows K=k (lanes0-15) / K=k+2
    b[0] = W[(long long)(k + 2 * half + 0) * OUT_C + n_base + l];
    b[1] = W[(long long)(k + 2 * half + 1) * OUT_C + n_base + l];
    c = __builtin_amdgcn_wmma_f32_16x16x4_f32(false, a, false, b,
                                              (short)0, c, false, false);
  }

  // C/D layout: lanes0-15 -> M=v, N=lane; lanes16-31 -> M=v+8, N=lane-16
  #pragma unroll
  for (int v = 0; v < 8; ++v) {
    int gr = m_base + v + 8 * half;
    if (gr < N) xw[(long long)gr * OUT_C + n_base + l] = c[v];
  }
}

// ---------------- per-(node,head) attention scores -----------------------
__global__ void gat_attn_scores(const float* __restrict__ xw,
                                const float* __restrict__ att_src,
                                const float* __restrict__ att_dst,
                                float* __restrict__ a_src,
                                float* __restrict__ a_dst, int N) {
  int i = blockIdx.x * blockDim.x + threadIdx.x;
  if (i >= N * HEADS) return;
  int n = i / HEADS, h = i % HEADS;
  const float* row = xw + (long long)n * OUT_C + h * HEAD_D;
  const float* vs  = att_src + h * HEAD_D;
  const float* vd  = att_dst + h * HEAD_D;
  float s = 0.f, d = 0.f;
  #pragma unroll
  for (int j = 0; j < HEAD_D; ++j) { s += row[j] * vs[j]; d += row[j] * vd[j]; }
  a_src[i] = s;
  a_dst[i] = d;
}

// order-preserving float <-> uint (for atomicMax over signed floats)
__device__ __forceinline__ unsigned f2o(float f) {
  unsigned u = __float_as_uint(f);
  return (u & 0x80000000u) ? ~u : (u | 0x80000000u);
}
__device__ __forceinline__ float o2f(unsigned u) {
  return (u & 0x80000000u) ? __uint_as_float(u & 0x7FFFFFFFu) : __uint_as_float(~u);
}

// ---------------- per-(edge,head): leaky_relu + segment max --------------
__global__ void gat_edge_alpha(const long long* __restrict__ ei,
                               const float* __restrict__ a_src,
                               const float* __restrict__ a_dst,
                               float* __restrict__ alpha_e,
                               unsigned* __restrict__ amax_u, long long E) {
  long long i = (long long)blockIdx.x * blockDim.x + threadIdx.x;
  if (i >= E * HEADS) return;
  long long e = i >> 2;
  int h = (int)(i & 3);
  int row = (int)ei[e];
  int col = (int)ei[E + e];
  float a = a_src[row * HEADS + h] + a_dst[col * HEADS + h];
  a = (a > 0.f) ? a : 0.2f * a;                  // leaky_relu(0.2)
  alpha_e[i] = a;
  atomicMax(&amax_u[row * HEADS + h], f2o(a));
}

// ---------------- finalize segment max (empty -> 0) ----------------------
__global__ void gat_finalize_amax(unsigned* __restrict__ amax_u, int N) {
  int i = blockIdx.x * blockDim.x + threadIdx.x;
  if (i >= N * HEADS) return;
  unsigned u = amax_u[i];
  float m = (u == 0u) ? 0.0f : o2f(u);
  if (!isfinite(m)) m = 0.0f;
  ((float*)amax_u)[i] = m;                       // in place: uint -> float
}

// ---------------- per-(edge,head): exp + segment sum ---------------------
__global__ void gat_edge_exp(const long long* __restrict__ ei,
                             float* __restrict__ alpha_e,
                             const float* __restrict__ amax,
                             float* __restrict__ asum, long long E) {
  long long i = (long long)blockIdx.x * blockDim.x + threadIdx.x;
  if (i >= E * HEADS) return;
  long long e = i >> 2;
  int h = (int)(i & 3);
  int row = (int)ei[e];
  float v = __expf(alpha_e[i] - amax[row * HEADS + h]);
  alpha_e[i] = v;
  atomicAdd(&asum[row * HEADS + h], v);
}

// ---------------- one wave per edge: normalized message scatter ----------
__global__ void gat_edge_scatter(const long long* __restrict__ ei,
                                 const float* __restrict__ alpha_e,
                                 const float* __restrict__ asum,
                                 const float* __restrict__ xw,
                                 float* __restrict__ out, long long E) {
  long long e = (long long)blockIdx.x * (blockDim.x / 32) + (threadIdx.x >> 5);
  if (e >= E) return;
  int lane = threadIdx.x & 31;
  int row = (int)ei[e];
  int col = (int)ei[E + e];
  const float* src = xw + (long long)col * OUT_C;
  float* dst = out + (long long)row * OUT_C;
  #pragma unroll
  for (int h = 0; h < HEADS; ++h) {
    float coef = alpha_e[e * HEADS + h] / (asum[row * HEADS + h] + 1e-16f);
    int idx = h * HEAD_D + lane;                 // coalesced 128-float row
    atomicAdd(&dst[idx], src[idx] * coef);
  }
}

// ------------------------------ launcher ---------------------------------
extern "C" void kernel_launch(void* const* d_in, const int* in_sizes, int n_in,
                              void* d_out, int out_size, void* d_ws, size_t ws_size,
                              hipStream_t stream) {
  const float*     x        = (const float*)d_in[0];
  const float*     W        = (const float*)d_in[1];
  const float*     att_src  = (const float*)d_in[2];
  const float*     att_dst  = (const float*)d_in[3];
  // d_in[4] = edge_weight: unused by the reference
  const long long* ei       = (const long long*)d_in[5];
  float*           out      = (float*)d_out;

  const int       N = in_sizes[0] / IN_C;     // 100000
  const long long E = (long long)in_sizes[4]; // 1600000

  // workspace carve-up (256B aligned)
  auto align = [](size_t v) { return (v + 255) & ~(size_t)255; };
  char* ws = (char*)d_ws;
  size_t off = 0;
  float*    xw      = (float*)(ws + off); off = align(off + (size_t)N * OUT_C * 4);
  float*    a_src_n = (float*)(ws + off); off = align(off + (size_t)N * HEADS * 4);
  float*    a_dst_n = (float*)(ws + off); off = align(off + (size_t)N * HEADS * 4);
  unsigned* amax_u  = (unsigned*)(ws + off); off = align(off + (size_t)N * HEADS * 4);
  float*    asum    = (float*)(ws + off); off = align(off + (size_t)N * HEADS * 4);
  float*    alpha_e = (float*)(ws + off); off = align(off + (size_t)E * HEADS * 4);
  (void)ws_size; (void)n_in; (void)out_size;

  // 1) zero accumulators
  gat_zero<<<2048, 256, 0, stream>>>(out, (long long)N * OUT_C,
                                     amax_u, (long long)N * HEADS,
                                     asum, (long long)N * HEADS);
  // 2) xw = x @ W (WMMA fp32)
  gat_gemm_xw<<<(N + 15) / 16, 256, 0, stream>>>(x, W, xw, N);
  // 3) attention scores
  gat_attn_scores<<<(N * HEADS + 255) / 256, 256, 0, stream>>>(xw, att_src, att_dst,
                                                               a_src_n, a_dst_n, N);
  // 4) per-edge alpha + segment max
  long long eh = E * HEADS;
  gat_edge_alpha<<<(unsigned)((eh + 255) / 256), 256, 0, stream>>>(ei, a_src_n, a_dst_n,
                                                                   alpha_e, amax_u, E);
  // 5) finalize max
  gat_finalize_amax<<<(N * HEADS + 255) / 256, 256, 0, stream>>>(amax_u, N);
  // 6) exp + segment sum
  gat_edge_exp<<<(unsigned)((eh + 255) / 256), 256, 0, stream>>>(ei, alpha_e,
                                                                 (const float*)amax_u, asum, E);
  // 7) normalized scatter (1 wave / edge, 8 edges / block)
  gat_edge_scatter<<<(unsigned)((E + 7) / 8), 256, 0, stream>>>(ei, alpha_e, asum,
                                                                xw, out, E);
}